// VectorQuantizer_69578470195285
// MI455X (gfx1250) — compile-verified
//
#include <hip/hip_runtime.h>
#include <hip/hip_bf16.h>
#include <math.h>

typedef float v2f __attribute__((ext_vector_type(2)));
typedef float v8f __attribute__((ext_vector_type(8)));

// Problem constants
#define TOK    32768      // 32*32*32 tokens
#define CDIM   256
#define KEMB   4096

// d_out layout (floats, concatenated in return order)
#define OFF_Q     ((size_t)0)          // [32,256,32,32] = 8388608
#define OFF_LOSS  ((size_t)8388608)
#define OFF_PERP  ((size_t)8388609)
#define OFF_ENC   ((size_t)8388610)    // [32768,4096]
#define OFF_IDX   ((size_t)142606338)  // [32,32,32]
#define OFF_DIST  ((size_t)142639106)  // [32768,4096]

// ---------------- init: e2 norms, zero histogram + loss accum ----------------
__global__ __launch_bounds__(256) void vq_init_kernel(
    const float* __restrict__ emb, float* __restrict__ e2ws,
    float* __restrict__ countsWs, float* __restrict__ lossWs)
{
    int k = blockIdx.x * 256 + threadIdx.x;      // 16 blocks * 256 = 4096
    const float* r = emb + (size_t)k * CDIM;
    float s = 0.f;
#pragma unroll 8
    for (int c = 0; c < CDIM; ++c) { float v = r[c]; s += v * v; }
    e2ws[k] = s;
    countsWs[k] = 0.f;
    if (k == 0) lossWs[0] = 0.f;
}

// ---------------- main: WMMA distance GEMM + argmin + all outputs ----------------
__global__ __launch_bounds__(256) void vq_main_kernel(
    const float* __restrict__ inp, const float* __restrict__ emb,
    const float* __restrict__ e2ws, float* __restrict__ countsWs,
    float* __restrict__ lossWs, float* __restrict__ qOut,
    float* __restrict__ encOut, float* __restrict__ idxOut,
    float* __restrict__ distOut)
{
    __shared__ float Alds[32 * CDIM];   // 32 token rows x 256 chans = 32KB
    __shared__ float x2lds[32];
    __shared__ float pMin[4][32];
    __shared__ int   pIdx[4][32];
    __shared__ int   fIdx[32];

    const int tid   = threadIdx.x;
    const int mbase = blockIdx.x * 32;            // token tile base (aligned to 32)
    const int nimg  = mbase >> 10;                // t = ((n*32 + w)*32 + h)
    const int wcol  = (mbase >> 5) & 31;          // tile spans h=0..31 at fixed (n,w)

    // Stage A tile: Alds[h][c] = inputs[n][c][h][w]   (inputs are NCHW)
    {
        const size_t gbase = (size_t)nimg * 262144 + (size_t)wcol;
        for (int idx = tid; idx < 32 * CDIM; idx += 256) {
            int h = idx & 31, c = idx >> 5;
            Alds[h * CDIM + c] = inp[gbase + (size_t)c * 1024 + (size_t)h * 32];
        }
    }
    __syncthreads();

    if (tid < 32) {                                // per-row ||x||^2
        const float* r = &Alds[tid * CDIM];
        float s = 0.f;
#pragma unroll 8
        for (int c = 0; c < CDIM; ++c) { float v = r[c]; s += v * v; }
        x2lds[tid] = s;
    }
    __syncthreads();

    const int lane   = tid & 31;
    const int wave   = tid >> 5;
    const int half   = lane >> 4;                  // K-pair select (ISA A 16x4 layout)
    const int n16    = lane & 15;
    const int rowgrp = (wave & 1) * 16;            // 2 rowgroups x 4 colgroups
    const int colgrp = wave >> 1;

    float xr[8];
#pragma unroll
    for (int r = 0; r < 8; ++r) xr[r] = x2lds[rowgrp + r + 8 * half];

    float minv[8]; int mini[8];
#pragma unroll
    for (int r = 0; r < 8; ++r) { minv[r] = 3.4e38f; mini[r] = 0; }

    const float* aptr = &Alds[(rowgrp + n16) * CDIM + 2 * half];

    for (int chunk = 0; chunk < 64; ++chunk) {     // sweep K=4096 in 64-col chunks
        const int col = chunk * 64 + colgrp * 16 + n16;
        const float* bptr = emb + (size_t)col * CDIM + 2 * half;
        v8f acc = {};
#pragma unroll
        for (int cc = 0; cc < CDIM; cc += 4) {
            v2f a = *(const v2f*)(aptr + cc);      // A[m][k], k = 2*half..+1
            v2f b = *(const v2f*)(bptr + cc);      // B[k][n] = E[col][c]
            acc = __builtin_amdgcn_wmma_f32_16x16x4_f32(
                false, a, false, b, (short)0, acc, false, false);
        }
        const float e2v = e2ws[col];
#pragma unroll
        for (int r = 0; r < 8; ++r) {              // C layout: VGPR r -> row r (+8 upper half)
            float dist = xr[r] + e2v - 2.0f * acc[r];
            int   trow = mbase + rowgrp + r + 8 * half;
            size_t off = (size_t)trow * KEMB + col;
            __builtin_nontemporal_store(dist, distOut + off);   // 512MB NT stream
            __builtin_nontemporal_store(0.0f, encOut + off);    // zero-fill one-hot
            if (dist < minv[r]) { minv[r] = dist; mini[r] = col; }
        }
    }

    // butterfly min across the 16 lanes holding each row (within half-wave)
#pragma unroll
    for (int r = 0; r < 8; ++r) {
        float v = minv[r]; int i = mini[r];
#pragma unroll
        for (int off = 1; off < 16; off <<= 1) {
            float ov = __shfl_xor(v, off, 32);
            int   oi = __shfl_xor(i, off, 32);
            if (ov < v || (ov == v && oi < i)) { v = ov; i = oi; }
        }
        if (n16 == r) {
            pMin[colgrp][rowgrp + r + 8 * half] = v;
            pIdx[colgrp][rowgrp + r + 8 * half] = i;
        }
    }

    // Drain encodings zero-fill stores before the one-hot scatter below.
    asm volatile("s_wait_storecnt 0x0" ::: "memory");
    __syncthreads();

    if (tid < 32) {                                // reduce over 4 colgroups
        float bv = pMin[0][tid]; int bi = pIdx[0][tid];
#pragma unroll
        for (int g = 1; g < 4; ++g) {
            float v = pMin[g][tid]; int i = pIdx[g][tid];
            if (v < bv || (v == bv && i < bi)) { bv = v; bi = i; }
        }
        fIdx[tid] = bi;
    }
    __syncthreads();

    if (tid < 32) {                                // indices, histogram, one-hot 1.0
        int t = mbase + tid, k = fIdx[tid];
        idxOut[t] = (float)k;
        atomicAdd(&countsWs[k], 1.0f);
        encOut[(size_t)t * KEMB + k] = 1.0f;
    }

    // quantized output (NCHW scatter) + loss partial
    {
        int row = tid >> 3, seg = tid & 7;
        int k = fIdx[row];
        int t = mbase + row;
        int h = t & 31;
        const float* erow = emb + (size_t)k * CDIM;
        float* qb = qOut + (size_t)nimg * 262144 + (size_t)h * 32 + wcol;
        float s = 0.f;
#pragma unroll 8
        for (int c = seg * 32; c < seg * 32 + 32; ++c) {
            float e = erow[c];
            float x = Alds[row * CDIM + c];
            float d = e - x;
            s += d * d;
            qb[(size_t)c * 1024] = e;
        }
#pragma unroll
        for (int off = 1; off < 32; off <<= 1) s += __shfl_xor(s, off, 32);
        if (lane == 0) atomicAdd(lossWs, s);
    }
}

// ---------------- finalize: perplexity + loss ----------------
__global__ __launch_bounds__(256) void vq_final_kernel(
    const float* __restrict__ countsWs, const float* __restrict__ lossWs,
    float* __restrict__ out)
{
    __shared__ float red[8];
    int tid = threadIdx.x;
    float s = 0.f;
#pragma unroll 4
    for (int j = 0; j < 16; ++j) {
        float p = countsWs[tid * 16 + j] * (1.0f / 32768.0f);
        s += p * logf(p + 1e-10f);
    }
#pragma unroll
    for (int off = 1; off < 32; off <<= 1) s += __shfl_xor(s, off, 32);
    if ((tid & 31) == 0) red[tid >> 5] = s;
    __syncthreads();
    if (tid == 0) {
        float tot = 0.f;
#pragma unroll
        for (int i = 0; i < 8; ++i) tot += red[i];
        out[OFF_PERP] = expf(-tot);
        // loss = q_latent + 0.25*e_latent = 1.25 * mean((q-x)^2); mean over T*C
        out[OFF_LOSS] = 1.25f * lossWs[0] / 8388608.0f;
    }
}

extern "C" void kernel_launch(void* const* d_in, const int* in_sizes, int n_in,
                              void* d_out, int out_size, void* d_ws, size_t ws_size,
                              hipStream_t stream)
{
    const float* inp = (const float*)d_in[0];    // [32,256,32,32]
    const float* emb = (const float*)d_in[1];    // [4096,256]
    float* out = (float*)d_out;
    float* ws  = (float*)d_ws;

    float* e2ws     = ws;           // 4096 floats
    float* countsWs = ws + 4096;    // 4096 floats
    float* lossWs   = ws + 8192;    // 1 float

    float* qOut    = out + OFF_Q;
    float* encOut  = out + OFF_ENC;
    float* idxOut  = out + OFF_IDX;
    float* distOut = out + OFF_DIST;

    vq_init_kernel<<<KEMB / 256, 256, 0, stream>>>(emb, e2ws, countsWs, lossWs);
    vq_main_kernel<<<TOK / 32, 256, 0, stream>>>(inp, emb, e2ws, countsWs, lossWs,
                                                 qOut, encOut, idxOut, distOut);
    vq_final_kernel<<<1, 256, 0, stream>>>(countsWs, lossWs, out);
}